// CropperQAT_15719580304238
// MI455X (gfx1250) — compile-verified
//
#include <hip/hip_runtime.h>
#include <hip/hip_bf16.h>
#include <stddef.h>

// ---------------------------------------------------------------------------
// RoIAlign (8x8, ratio=1, 3 pyramid levels, fake-quant ROIs) as WMMA GEMMs.
//
// Per (roi, level, channel-pair):  out[16x16] = Wy[16x16] x Q[16x16]
//   Wy[j, 2j]   = (1-fy_j)*valid_y_j      (rows 8..15 zero-padded)
//   Wy[j, 2j+1] =    fy_j *valid_y_j
//   Q[r, n]     = x-lerped gathered feature row (r even -> yl, odd -> yh),
//                 n = chanSel*8 + x, x-lerp weights carry valid_x mask.
// K=16 = chain of four V_WMMA_F32_16X16X4_F32 accumulations.
//
// Roofline: the whole feature pyramid (~88 MB) fits in the 192 MB L2, so the
// gathers stay on-chip; HBM traffic = 201 MB of output stores (~9 us @
// 23.3 TB/s).  Therefore:
//   * the WMMA reduction is free and keeps VALU for addressing,
//   * stores are made fully coalesced via a per-wave LDS transpose of the
//     D-matrix (one contiguous global_store_b128 per lane, full wave active),
//   * all ROI/weight/offset math is hoisted to the level loop; the inner
//     channel-pair loop is just 16 gathers + 8 FMAs + 4 WMMAs + 1 b128 store.
// ---------------------------------------------------------------------------

typedef float v2f __attribute__((ext_vector_type(2)));
typedef float v4f __attribute__((ext_vector_type(4)));
typedef float v8f __attribute__((ext_vector_type(8)));

__device__ __forceinline__ float fq(float v) {
    // fake_quant: clip(round(v/0.25), -32768, 32767) * 0.25 ; RNE like jnp.round
    float q = rintf(v * 4.0f);
    q = fminf(fmaxf(q, -32768.0f), 32767.0f);
    return q * 0.25f;
}

// torchvision-style coordinate prep (aligned=False). Indices always in [0,L-1].
__device__ __forceinline__ void prep_axis(float c, int L,
                                          int& lo, int& hi, float& fr, float& valid) {
    valid = (c >= -1.0f && c <= (float)L) ? 1.0f : 0.0f;
    c = fmaxf(c, 0.0f);
    int l0 = (int)floorf(c);
    bool he = l0 >= (L - 1);
    lo = he ? (L - 1) : l0;
    hi = he ? (L - 1) : (l0 + 1);
    float cc = he ? (float)(L - 1) : c;
    fr = cc - (float)lo;
}

__global__ void __launch_bounds__(256)
roialign_qat_wmma(const float* __restrict__ f0,
                  const float* __restrict__ f1,
                  const float* __restrict__ f2,
                  const float* __restrict__ pixel,
                  const int*   __restrict__ bidx,
                  float* __restrict__ out,
                  int nroi) {
    __shared__ float lds[8][128];          // per-wave D-matrix transpose tile

    const int a = blockIdx.x;
    if (a >= nroi) return;                  // uniform per block (EXEC stays full)

    const int lane = threadIdx.x & 31;
    const int wave = threadIdx.x >> 5;
    const int hi16 = lane >> 4;             // 0: lanes 0-15, 1: lanes 16-31
    const int lm   = lane & 15;             // M for A-matrix, N for B/D matrices

    const float px = pixel[2 * a + 0];
    const float py = pixel[2 * a + 1];
    const int   b  = bidx[a];

#pragma unroll
    for (int level = 0; level < 3; ++level) {
        float inv_s; int L; const float* f;
        if (level == 0)      { inv_s = 0.25f;   L = 256; f = f0; }
        else if (level == 1) { inv_s = 0.125f;  L = 128; f = f1; }
        else                 { inv_s = 0.0625f; L = 64;  f = f2; }

        // ---- fake-quantized ROI box (exact pow2 scales) ----
        const float x1 = fq(fmaxf(px * inv_s - 4.0f, 0.0f));
        const float y1 = fq(fmaxf(py * inv_s - 4.0f, 0.0f));
        const float x2 = fq(fmaxf(px * inv_s + 4.0f, 0.0f));
        const float y2 = fq(fmaxf(py * inv_s + 4.0f, 0.0f));
        const float bw = fmaxf(x2 - x1, 1.0f) * 0.125f;   // roi_w / 8
        const float bh = fmaxf(y2 - y1, 1.0f) * 0.125f;   // roi_h / 8

        // ---- per-lane x sample for B-matrix column n = lm (channel-independent)
        const int xo = lm & 7;              // output x
        const int cs = lm >> 3;             // which of the 2 packed channels
        int xl, xh; float fx, vx;
        prep_axis(x1 + ((float)xo + 0.5f) * bw, L, xl, xh, fx, vx);
        const float wx0 = (1.0f - fx) * vx;
        const float wx1 = fx * vx;

        // ---- per-lane y weights for A-matrix row M = lm (rows 8..15 zero) ----
        int t0, t1; float Afy, Avy;
        prep_axis(y1 + ((float)lm + 0.5f) * bh, L, t0, t1, Afy, Avy);
        const float w0 = (lm < 8) ? (1.0f - Afy) * Avy : 0.0f;
        const float w1 = (lm < 8) ? Afy * Avy : 0.0f;

        // ---- hoisted per-K-step A operands and gather offsets ----
        v2f am[4];
        int o00[4], o01[4], o10[4], o11[4];   // plane-relative gather offsets
#pragma unroll
        for (int k = 0; k < 4; ++k) {
            // B rows for this K-step: lanes 0-15 own j=2k, lanes 16-31 j=2k+1
            const int j = 2 * k + hi16;
            int yl, yh; float fy, vy;
            prep_axis(y1 + ((float)j + 0.5f) * bh, L, yl, yh, fy, vy);
            (void)fy; (void)vy;               // vy is applied on the A side
            o00[k] = yl * L + xl;  o01[k] = yl * L + xh;
            o10[k] = yh * L + xl;  o11[k] = yh * L + xh;

            // A: lane holds Wy[lm, rA0] in v0 and Wy[lm, rA0+1] in v1
            const int rA0 = 4 * k + 2 * hi16;
            am[k].x = (rA0     == 2 * lm) ? w0 : ((rA0     == 2 * lm + 1) ? w1 : 0.0f);
            am[k].y = (rA0 + 1 == 2 * lm) ? w0 : ((rA0 + 1 == 2 * lm + 1) ? w1 : 0.0f);
        }

        // ---- inner loop: 4 channel-pairs per wave (32 pairs / 8 waves) ----
        for (int pair = wave; pair < 32; pair += 8) {
            const int c0 = 2 * pair;
            const float* fb = f + ((size_t)b * 64 + (size_t)(c0 + cs))
                                  * (size_t)L * (size_t)L;

            v8f acc = {0.f, 0.f, 0.f, 0.f, 0.f, 0.f, 0.f, 0.f};
#pragma unroll
            for (int k = 0; k < 4; ++k) {
                v2f bm;
                bm.x = wx0 * fb[o00[k]] + wx1 * fb[o01[k]];   // Q[4k+2*hi16  , n]
                bm.y = wx0 * fb[o10[k]] + wx1 * fb[o11[k]];   // Q[4k+2*hi16+1, n]
                acc = __builtin_amdgcn_wmma_f32_16x16x4_f32(
                          false, am[k], false, bm, (short)0, acc, false, false);
            }

            // ---- LDS transpose: D rows (VGPRs of lanes 0-15) -> contiguous ----
            // lds layout: [cs*64 + y*8 + x]  == output order for the 2 channels
            if (hi16 == 0) {
#pragma unroll
                for (int y = 0; y < 8; ++y)
                    lds[wave][cs * 64 + y * 8 + xo] = acc[y];
            }
            // Same-wave DS ops are in-order: RAW through LDS is safe w/o barrier.
            const v4f vals = *(const v4f*)&lds[wave][4 * lane];

            // one fully-coalesced 512B store per wave (b128 per lane)
            float* o = out + ((size_t)a * 192 + (size_t)(level * 64 + c0)) * 64
                           + (size_t)(4 * lane);
            *(v4f*)o = vals;
        }
    }
}

extern "C" void kernel_launch(void* const* d_in, const int* in_sizes, int n_in,
                              void* d_out, int out_size, void* d_ws, size_t ws_size,
                              hipStream_t stream) {
    (void)n_in; (void)out_size; (void)d_ws; (void)ws_size;
    const float* f0    = (const float*)d_in[0];   // [4,64,256,256]
    const float* f1    = (const float*)d_in[1];   // [4,64,128,128]
    const float* f2    = (const float*)d_in[2];   // [4,64,64,64]
    const float* pixel = (const float*)d_in[3];   // [A,2]
    const int*   bi    = (const int*)  d_in[4];   // [A]
    float*       out   = (float*)d_out;           // [A,192,8,8]

    const int nroi = in_sizes[3] / 2;             // 4096
    roialign_qat_wmma<<<nroi, 256, 0, stream>>>(f0, f1, f2, pixel, bi, out, nroi);
}